// BiLSTM_CRF_88871463289215
// MI455X (gfx1250) — compile-verified
//
#include <hip/hip_runtime.h>
#include <hip/hip_bf16.h>

// ---- problem constants (match reference) ----
#define LSEQ   8192
#define EMB    256
#define HDIM   256           // per-direction hidden
#define GDIM   1024          // 4*HDIM gates
#define HFULL  512
#define NT     34            // TAGS + START + STOP
#define TSTART 32
#define FNEG   (-10000.0f)

#define XS_STRIDE 260        // 16x260 f32 A-tile in LDS: 16B-aligned rows, conflict-free
#define NCACHE   256         // Whh rows cached in LDS (gate i), 8 full waves
#define WS_STRIDE 258        // cached-row stride (8B aligned, conflict-free b64 reads)
#define LSTM_DYN_BYTES ((size_t)NCACHE * WS_STRIDE * 4)

typedef __attribute__((ext_vector_type(2))) float v2f;
typedef __attribute__((ext_vector_type(8))) float v8f;

__device__ __forceinline__ float sigm(float x) {
    return 1.0f / (1.0f + __expf(-x));
}

// Async global->LDS helpers (gfx1250 VGLOBAL async ops, tracked by ASYNCcnt).
// lp = LDS byte offset (low 32 bits of a generic shared pointer), gp = global ptr.
__device__ __forceinline__ void async_g2l_b128(unsigned lp, const void* gp) {
    asm volatile("global_load_async_to_lds_b128 %0, %1, off"
                 :: "v"(lp), "v"(gp) : "memory");
}
__device__ __forceinline__ void async_g2l_b64(unsigned lp, const void* gp) {
    asm volatile("global_load_async_to_lds_b64 %0, %1, off"
                 :: "v"(lp), "v"(gp) : "memory");
}
__device__ __forceinline__ void wait_async0() {
    asm volatile("s_wait_asynccnt 0" ::: "memory");
}

// ---------------------------------------------------------------------------
// K1: embedding gather  X[l, :] = embed[words[l], :]   (float4 vectorized)
// ---------------------------------------------------------------------------
__global__ __launch_bounds__(256) void k_gather(const int* __restrict__ words,
                                                const float* __restrict__ embed,
                                                float* __restrict__ X) {
    int idx = blockIdx.x * 256 + threadIdx.x;          // over LSEQ*EMB/4
    if (idx >= LSEQ * (EMB / 4)) return;
    int row  = idx / (EMB / 4);
    int col4 = idx % (EMB / 4);
    const float4* src = (const float4*)(embed + (size_t)words[row] * EMB);
    ((float4*)(X + (size_t)row * EMB))[col4] = src[col4];
}

// ---------------------------------------------------------------------------
// K2: input projection GEMM via f32 WMMA:  G = X @ W^T + b
//     One block per 16-row M-tile; A tile staged in LDS via async-to-LDS;
//     8 waves each compute 8 of the 64 N-tiles (v_wmma_f32_16x16x4_f32).
// ---------------------------------------------------------------------------
__global__ __launch_bounds__(256) void k_gemm_inproj(const float* __restrict__ X,
                                                     const float* __restrict__ W,
                                                     const float* __restrict__ bias,
                                                     float* __restrict__ G) {
    __shared__ float Xs[16][XS_STRIDE];
    const int tid = threadIdx.x;
    const int m0  = blockIdx.x << 4;

    // stage A tile (16 x 256 f32) into LDS: 1024 b128 transfers, 4 per thread
    #pragma unroll
    for (int i = 0; i < 4; ++i) {
        int e    = tid + (i << 8);          // 0..1023
        int row  = e >> 6;
        int c4   = e & 63;
        const float* gp = X + (size_t)(m0 + row) * EMB + (c4 << 2);
        unsigned lp = (unsigned)(uintptr_t)&Xs[row][c4 << 2];
        async_g2l_b128(lp, gp);
    }
    wait_async0();
    __syncthreads();

    const int wave = tid >> 5;              // 0..7
    const int lane = tid & 31;
    const int half = lane >> 4;             // 0: K=0,1   1: K=2,3
    const int l    = lane & 15;
    const float* arow = &Xs[l][2 * half];

    for (int j = 0; j < 8; ++j) {
        const int n0 = ((wave << 3) + j) << 4;
        const float* wrow = W + (size_t)(n0 + l) * EMB + 2 * half;

        v8f acc = {0.f, 0.f, 0.f, 0.f, 0.f, 0.f, 0.f, 0.f};
        #pragma unroll 8
        for (int k = 0; k < EMB; k += 4) {
            float2 av = *(const float2*)(arow + k);     // LDS
            float2 bv = *(const float2*)(wrow + k);     // global (L2-resident)
            v2f a; a.x = av.x; a.y = av.y;
            v2f b; b.x = bv.x; b.y = bv.y;
            acc = __builtin_amdgcn_wmma_f32_16x16x4_f32(
                false, a, false, b, (short)0, acc, false, false);
        }
        const int   n  = n0 + l;
        const float bn = bias[n];
        #pragma unroll
        for (int r = 0; r < 8; ++r) {
            int m = m0 + r + 8 * half;
            G[(size_t)m * GDIM + n] = acc[r] + bn;
        }
    }
}

// ---------------------------------------------------------------------------
// K3: persistent LSTM recurrence, one 1024-thread block per direction.
//     Thread tid owns gate-row tid of Whh (gate = tid>>8, unit = tid&255).
//     Gate-i rows (0..255) cached in LDS (async prefill); rest stream from L2.
//     h lives in LDS; c in registers of lanes 0..255.
// ---------------------------------------------------------------------------
__global__ __launch_bounds__(1024) void k_lstm(const float* __restrict__ Gf,
                                               const float* __restrict__ Gb,
                                               const float* __restrict__ Whh_f,
                                               const float* __restrict__ Whh_b,
                                               float* __restrict__ hf,
                                               float* __restrict__ hb) {
    extern __shared__ float wc[];           // NCACHE rows, stride WS_STRIDE
    __shared__ float sh[HDIM];              // hidden state
    __shared__ float sg[GDIM];              // gate pre-activations

    const bool bwd = (blockIdx.x == 1);
    const float* Gm  = bwd ? Gb    : Gf;
    const float* Whh = bwd ? Whh_b : Whh_f;
    float*       hs  = bwd ? hb    : hf;

    const int tid = threadIdx.x;            // 0..1023

    // async prefill of cached Whh rows: NCACHE x 128 b64 transfers
    for (int e = tid; e < NCACHE * (HDIM / 2); e += 1024) {
        int row = e >> 7;                   // /128
        int c2  = e & 127;
        const float* gp = Whh + (size_t)row * HDIM + 2 * c2;
        unsigned lp = (unsigned)(uintptr_t)&wc[row * WS_STRIDE + 2 * c2];
        async_g2l_b64(lp, gp);
    }
    wait_async0();

    float c = 0.0f;
    if (tid < HDIM) sh[tid] = 0.0f;
    __syncthreads();

    const float4* Wg4 = (const float4*)(Whh + (size_t)tid * HDIM);  // streamed path
    const float2* Wc2 = (const float2*)&wc[(size_t)tid * WS_STRIDE]; // cached path

    for (int s = 0; s < LSEQ; ++s) {
        const int step = bwd ? (LSEQ - 1 - s) : s;
        const int nstep = bwd ? (step - 1) : (step + 1);
        if (nstep >= 0 && nstep < LSEQ)     // prefetch next step's gate row
            __builtin_prefetch(&Gm[(size_t)nstep * GDIM + tid], 0, 1);

        float acc = Gm[(size_t)step * GDIM + tid];

        if (tid < NCACHE) {                 // waves 0..7: Whh from LDS
            const float2* hv2 = (const float2*)sh;
            #pragma unroll 16
            for (int k = 0; k < HDIM / 2; ++k) {
                float2 w = Wc2[k];
                float2 h2 = hv2[k];
                acc += w.x * h2.x + w.y * h2.y;
            }
        } else {                            // waves 8..31: Whh from L2
            const float4* hv4 = (const float4*)sh;
            #pragma unroll 8
            for (int k = 0; k < HDIM / 4; ++k) {
                float4 w = Wg4[k];
                float4 h4 = hv4[k];
                acc += w.x * h4.x + w.y * h4.y + w.z * h4.z + w.w * h4.w;
            }
        }
        sg[tid] = acc;
        __syncthreads();                    // gates ready; all reads of old h done

        if (tid < HDIM) {
            float ig = sigm(sg[tid]);
            float fg = sigm(sg[HDIM + tid]);
            float gg = tanhf(sg[2 * HDIM + tid]);
            float og = sigm(sg[3 * HDIM + tid]);
            c = fg * c + ig * gg;
            float hnew = og * tanhf(c);
            hs[(size_t)step * HDIM + tid] = hnew;
            sh[tid] = hnew;
        }
        __syncthreads();                    // new h visible
    }
}

// ---------------------------------------------------------------------------
// K4: output projection via f32 WMMA: feats = [hf|hb] @ Wout^T + bout
//     M = LSEQ, N = NT (padded to 48 -> 3 tiles), K = HFULL (256 hf + 256 hb)
// ---------------------------------------------------------------------------
__global__ __launch_bounds__(256) void k_outproj(const float* __restrict__ hf,
                                                 const float* __restrict__ hb,
                                                 const float* __restrict__ Wout,
                                                 const float* __restrict__ bout,
                                                 float* __restrict__ feats) {
    const int gtid = blockIdx.x * 256 + threadIdx.x;
    const int wave = gtid >> 5;
    const int lane = threadIdx.x & 31;
    const int tm   = wave / 3;
    const int tn   = wave % 3;
    const int m0   = tm << 4;
    const int n0   = tn << 4;
    const int half = lane >> 4;
    const int l    = lane & 15;
    const int n    = n0 + l;
    const bool nok = (n < NT);
    const float* wrow = nok ? (Wout + (size_t)n * HFULL) : Wout;

    v8f acc = {0.f, 0.f, 0.f, 0.f, 0.f, 0.f, 0.f, 0.f};
    #pragma unroll 4
    for (int k = 0; k < HFULL; k += 4) {
        const float* hsrc = (k < HDIM)
            ? (hf + (size_t)(m0 + l) * HDIM + k)
            : (hb + (size_t)(m0 + l) * HDIM + (k - HDIM));
        float2 av = *(const float2*)(hsrc + 2 * half);
        v2f a; a.x = av.x; a.y = av.y;
        float2 bv = *(const float2*)(wrow + k + 2 * half);
        v2f b;
        b.x = nok ? bv.x : 0.0f;
        b.y = nok ? bv.y : 0.0f;
        acc = __builtin_amdgcn_wmma_f32_16x16x4_f32(
            false, a, false, b, (short)0, acc, false, false);
    }
    if (nok) {
        const float bn = bout[n];
        #pragma unroll
        for (int r = 0; r < 8; ++r) {
            int m = m0 + r + 8 * half;
            feats[(size_t)m * NT + n] = acc[r] + bn;
        }
    }
}

// ---------------------------------------------------------------------------
// K5: CRF forward (block 0) and backward (block 1), 34 active lanes.
// ---------------------------------------------------------------------------
__global__ __launch_bounds__(64) void k_crf(const float* __restrict__ feats,
                                            const float* __restrict__ trans,
                                            float* __restrict__ la,
                                            float* __restrict__ lb) {
    const int j = threadIdx.x;
    __shared__ float st[NT];

    if (blockIdx.x == 0) {
        // forward: alpha_t[j] = lse_i(alpha[i] + trans[j][i]) + feat_t[j]
        if (j < NT) st[j] = (j == TSTART) ? 0.0f : FNEG;
        __syncthreads();
        for (int t = 0; t < LSEQ; ++t) {
            float a = 0.0f;
            if (j < NT) {
                const float* tr = trans + (size_t)j * NT;
                float m = -3.4e38f;
                #pragma unroll
                for (int i = 0; i < NT; ++i)
                    m = fmaxf(m, st[i] + tr[i]);
                float s = 0.0f;
                #pragma unroll
                for (int i = 0; i < NT; ++i)
                    s += __expf(st[i] + tr[i] - m);
                a = m + __logf(s) + feats[(size_t)t * NT + j];
            }
            __syncthreads();
            if (j < NT) { st[j] = a; la[(size_t)t * NT + j] = a; }
            __syncthreads();
        }
    } else {
        // backward: beta_L[j] = lse_i(trans[i][j]);
        // beta_t[j] = lse_i(beta_{t+1}[i] + feat_{t+1}[i] + trans[i][j])
        if (j < NT) {
            float m = -3.4e38f;
            #pragma unroll
            for (int i = 0; i < NT; ++i) m = fmaxf(m, trans[(size_t)i * NT + j]);
            float s = 0.0f;
            #pragma unroll
            for (int i = 0; i < NT; ++i) s += __expf(trans[(size_t)i * NT + j] - m);
            float b0 = m + __logf(s);
            st[j] = b0;
            lb[(size_t)(LSEQ - 1) * NT + j] = b0;
        }
        __syncthreads();
        for (int t = LSEQ - 2; t >= 0; --t) {
            float b = 0.0f;
            if (j < NT) {
                const float* fn = feats + (size_t)(t + 1) * NT;
                float m = -3.4e38f;
                #pragma unroll
                for (int i = 0; i < NT; ++i)
                    m = fmaxf(m, st[i] + fn[i] + trans[(size_t)i * NT + j]);
                float s = 0.0f;
                #pragma unroll
                for (int i = 0; i < NT; ++i)
                    s += __expf(st[i] + fn[i] + trans[(size_t)i * NT + j] - m);
                b = m + __logf(s);
            }
            __syncthreads();
            if (j < NT) { st[j] = b; lb[(size_t)t * NT + j] = b; }
            __syncthreads();
        }
    }
}

// ---------------------------------------------------------------------------
// K6: score = log_alpha + log_beta; tags = argmax (first max, like jnp.argmax)
//     d_out = [score (LSEQ*NT floats) | tags (LSEQ floats)]
// ---------------------------------------------------------------------------
__global__ __launch_bounds__(256) void k_score(const float* __restrict__ la,
                                               const float* __restrict__ lb,
                                               float* __restrict__ out) {
    int t = blockIdx.x * 256 + threadIdx.x;
    if (t >= LSEQ) return;
    float best = -3.4e38f;
    int   bi   = 0;
    for (int jj = 0; jj < NT; ++jj) {
        float v = la[(size_t)t * NT + jj] + lb[(size_t)t * NT + jj];
        out[(size_t)t * NT + jj] = v;
        if (v > best) { best = v; bi = jj; }
    }
    out[(size_t)LSEQ * NT + t] = (float)bi;
}

// ---------------------------------------------------------------------------
extern "C" void kernel_launch(void* const* d_in, const int* in_sizes, int n_in,
                              void* d_out, int out_size, void* d_ws, size_t ws_size,
                              hipStream_t stream) {
    const int*   words = (const int*)d_in[0];
    const float* embed = (const float*)d_in[1];
    const float* Wih_f = (const float*)d_in[2];
    const float* Whh_f = (const float*)d_in[3];
    const float* b_f   = (const float*)d_in[4];
    const float* Wih_b = (const float*)d_in[5];
    const float* Whh_b = (const float*)d_in[6];
    const float* b_b   = (const float*)d_in[7];
    const float* Wout  = (const float*)d_in[8];
    const float* bout  = (const float*)d_in[9];
    const float* trans = (const float*)d_in[10];

    float* ws    = (float*)d_ws;
    float* X     = ws;
    float* Gf    = X  + (size_t)LSEQ * EMB;
    float* Gb    = Gf + (size_t)LSEQ * GDIM;
    float* hf    = Gb + (size_t)LSEQ * GDIM;
    float* hb    = hf + (size_t)LSEQ * HDIM;
    float* feats = hb + (size_t)LSEQ * HDIM;
    float* la    = feats + (size_t)LSEQ * NT;
    float* lb    = la + (size_t)LSEQ * NT;
    float* out   = (float*)d_out;

    // allow >64KB dynamic LDS for the Whh cache (CDNA5: 320KB per workgroup)
    (void)hipFuncSetAttribute((const void*)k_lstm,
                              hipFuncAttributeMaxDynamicSharedMemorySize,
                              (int)LSTM_DYN_BYTES);

    // 1) embedding gather
    k_gather<<<(LSEQ * (EMB / 4) + 255) / 256, 256, 0, stream>>>(words, embed, X);

    // 2) input projections (WMMA GEMMs + async A staging), both directions
    k_gemm_inproj<<<LSEQ / 16, 256, 0, stream>>>(X, Wih_f, b_f, Gf);
    k_gemm_inproj<<<LSEQ / 16, 256, 0, stream>>>(X, Wih_b, b_b, Gb);

    // 3) persistent LSTM recurrence, one block per direction
    k_lstm<<<2, 1024, LSTM_DYN_BYTES, stream>>>(Gf, Gb, Whh_f, Whh_b, hf, hb);

    // 4) output projection (WMMA)
    k_outproj<<<(LSEQ / 16) * 3 / 8, 256, 0, stream>>>(hf, hb, Wout, bout, feats);

    // 5) CRF forward + backward (concurrent blocks)
    k_crf<<<2, 64, 0, stream>>>(feats, trans, la, lb);

    // 6) score + argmax -> d_out
    k_score<<<LSEQ / 256, 256, 0, stream>>>(la, lb, out);
}